// DynamicMaskHead_18202071400496
// MI455X (gfx1250) — compile-verified
//
#include <hip/hip_runtime.h>

typedef float v2f __attribute__((ext_vector_type(2)));
typedef float v8f __attribute__((ext_vector_type(8)));

#define NINST 128
#define CIN   8
#define H     128
#define W     192
#define HW    (H * W)
// param layout per instance (169 floats)
#define OFF_W1 0     // 8 x 10
#define OFF_W2 80    // 8 x 8
#define OFF_W3 144   // 8
#define OFF_B1 152   // 8
#define OFF_B2 160   // 8
#define OFF_B3 168   // 1
#define NPARAM 169

__device__ __forceinline__ float laneShfl(float v, int srcLane) {
    return __shfl(v, srcLane, 32);
}

// One block per (instance, segment). 8 waves/block. Each wave computes
// 16-pixel groups with f32 WMMA chains (exact fp32 math).
__global__ __launch_bounds__(256)
void dynmask_mlp_kernel(const float* __restrict__ feats,   // (2, 8, 128, 192)
                        const float* __restrict__ params,  // (128, 169)
                        const float* __restrict__ iloc,    // (128, 2)
                        const int*   __restrict__ im_inds, // (128)
                        const int*   __restrict__ fpn,     // (128)
                        float* __restrict__ logits)        // ws: (128, 24576)
{
    const int inst   = blockIdx.x;
    const int lane   = threadIdx.x & 31;
    const int waveId = threadIdx.x >> 5;
    const int m      = lane & 15;   // matrix row (A) / pixel (B,D)
    const int half   = lane >> 4;   // K-pair selector for A/B tiles
    const int kh     = half * 2;

    const float* p = params + inst * NPARAM;

    // ---- A tiles (weights), ISA 16x4 f32 layout:
    // lanes 0-15: v0=A[m][kb+0], v1=A[m][kb+1]; lanes 16-31: kb+2, kb+3
    v2f a1[3], a2[2];
#pragma unroll
    for (int t = 0; t < 3; ++t) {           // layer 1: K = 10 padded to 12
        int k0 = t * 4 + kh;
        float v0 = 0.f, v1 = 0.f;
        if (m < 8) {
            if (k0 + 0 < 10) v0 = p[OFF_W1 + m * 10 + k0 + 0];
            if (k0 + 1 < 10) v1 = p[OFF_W1 + m * 10 + k0 + 1];
        }
        a1[t] = v2f{v0, v1};
    }
#pragma unroll
    for (int t = 0; t < 2; ++t) {           // layer 2: K = 8
        int k0 = t * 4 + kh;
        float v0 = 0.f, v1 = 0.f;
        if (m < 8) {
            v0 = p[OFF_W2 + m * 8 + k0 + 0];
            v1 = p[OFF_W2 + m * 8 + k0 + 1];
        }
        a2[t] = v2f{v0, v1};
    }

    float w3[8], b1[8], b2[8];
#pragma unroll
    for (int j = 0; j < 8; ++j) {
        w3[j] = p[OFF_W3 + j];
        b1[j] = p[OFF_B1 + j];
        b2[j] = p[OFF_B2 + j];
    }
    const float b3 = p[OFF_B3];

    const int   im  = im_inds[inst];
    int lvl = fpn[inst]; lvl = lvl < 0 ? 0 : (lvl > 4 ? 4 : lvl);
    const float soiTab[5] = {64.f, 128.f, 256.f, 512.f, 1024.f};
    const float inv_soi = 1.0f / soiTab[lvl];
    const float lx = iloc[inst * 2 + 0];
    const float ly = iloc[inst * 2 + 1];

    const float* fbase = feats + (size_t)im * CIN * HW;

    const int nGroups    = HW / 16;                          // 1536
    const int wavesTotal = gridDim.y * (blockDim.x >> 5);    // 96
    const int gwave      = blockIdx.y * (blockDim.x >> 5) + waveId;

    for (int g = gwave; g < nGroups; g += wavesTotal) {
        const int p0  = g * 16;                 // group is within one row (192 % 16 == 0)
        const int pix = p0 + m;
        const int y   = pix / W;
        const int x   = pix - y * W;
        const float* fp = fbase + y * W + x;

        const float relx = (lx - (float)(x * 8 + 4)) * inv_soi;
        const float rely = (ly - (float)(y * 8 + 4)) * inv_soi;

        // Each lane needs 4 feature channels:
        //   half0 -> ch {2,3,6,7}, half1 -> ch {0,1,4,5}
        const int cA = half ? 0 : 2;
        const float f0 = fp[(size_t)(cA + 0) * HW];
        const float f1 = fp[(size_t)(cA + 1) * HW];
        const float f2 = fp[(size_t)(cA + 4) * HW];
        const float f3 = fp[(size_t)(cA + 5) * HW];

        // B tiles, layer 1 (K rows: {relx,rely,ch0..ch7,0,0}):
        v2f bt0 = half ? v2f{f0, f1} : v2f{relx, rely};  // K 0..3
        v2f bt1 = half ? v2f{f2, f3} : v2f{f0, f1};      // K 4..7
        v2f bt2 = half ? v2f{0.f, 0.f} : v2f{f2, f3};    // K 8..11 (10,11 pad)

        // ---- layer 1: D = W1 * X  (3 chained f32 WMMAs)
        v8f c = {};
        c = __builtin_amdgcn_wmma_f32_16x16x4_f32(false, a1[0], false, bt0, (short)0, c, false, false);
        c = __builtin_amdgcn_wmma_f32_16x16x4_f32(false, a1[1], false, bt1, (short)0, c, false, false);
        c = __builtin_amdgcn_wmma_f32_16x16x4_f32(false, a1[2], false, bt2, (short)0, c, false, false);

        float d[8];
#pragma unroll
        for (int j = 0; j < 8; ++j) {
            const float bj = half ? 0.f : b1[j];   // rows 8..15 are padding
            d[j] = fmaxf(c[j] + bj, 0.f);
        }

        // ---- rebuild B tiles for layer 2 from D layout
        // D: VGPR j, lanes 0-15 = channel j, pixel = lane
        // B: lanes<16 need ch {kb, kb+1}; lanes>=16 need ch {kb+2, kb+3} at pixel lane&15
        const int src = lane & 15;
        const float s2 = laneShfl(d[2], src);
        const float s3 = laneShfl(d[3], src);
        const float s6 = laneShfl(d[6], src);
        const float s7 = laneShfl(d[7], src);
        v2f B20 = v2f{half ? s2 : d[0], half ? s3 : d[1]};   // K 0..3
        v2f B21 = v2f{half ? s6 : d[4], half ? s7 : d[5]};   // K 4..7

        // ---- layer 2 (2 chained f32 WMMAs)
        v8f e = {};
        e = __builtin_amdgcn_wmma_f32_16x16x4_f32(false, a2[0], false, B20, (short)0, e, false, false);
        e = __builtin_amdgcn_wmma_f32_16x16x4_f32(false, a2[1], false, B21, (short)0, e, false, false);

        // ---- layer 3: 1 output channel -> per-lane dot over 8 accumulators
        float acc = b3;
#pragma unroll
        for (int j = 0; j < 8; ++j) {
            const float hj = fmaxf(e[j] + (half ? 0.f : b2[j]), 0.f);
            acc = fmaf(w3[j], hj, acc);
        }

        if (half == 0) {
            logits[(size_t)inst * HW + pix] = acc;
        }
    }
}

// aligned_bilinear(t, 2) reduced to a direct gather:
// out[oy][ox] = bilerp over i=max(oy-1,0), j=max(ox-1,0):
//   y0=i>>1, fy=0.5*(i&1), y1=min(y0+1,H-1); x likewise.
__global__ __launch_bounds__(256)
void dynmask_upsample_kernel(const float* __restrict__ logits,  // (128, 128, 192)
                             float* __restrict__ out)           // (128, 1, 256, 384)
{
    const int OW = 2 * W, OH = 2 * H;
    const size_t total = (size_t)NINST * OH * OW;
    size_t idx = (size_t)blockIdx.x * blockDim.x + threadIdx.x;
    if (idx >= total) return;

    const int ox = (int)(idx % OW);
    const size_t t = idx / OW;
    const int oy   = (int)(t % OH);
    const int inst = (int)(t / OH);

    const int i  = oy > 0 ? oy - 1 : 0;
    const int j  = ox > 0 ? ox - 1 : 0;
    const int y0 = i >> 1, x0 = j >> 1;
    const float fy = 0.5f * (float)(i & 1);
    const float fx = 0.5f * (float)(j & 1);
    const int y1 = (y0 + 1 < H) ? y0 + 1 : H - 1;
    const int x1 = (x0 + 1 < W) ? x0 + 1 : W - 1;

    const float* L = logits + (size_t)inst * HW;
    const float v00 = L[y0 * W + x0];
    const float v01 = L[y0 * W + x1];
    const float v10 = L[y1 * W + x0];
    const float v11 = L[y1 * W + x1];

    const float top = v00 * (1.f - fx) + v01 * fx;
    const float bot = v10 * (1.f - fx) + v11 * fx;
    out[idx] = top * (1.f - fy) + bot * fy;
}

extern "C" void kernel_launch(void* const* d_in, const int* in_sizes, int n_in,
                              void* d_out, int out_size, void* d_ws, size_t ws_size,
                              hipStream_t stream) {
    (void)in_sizes; (void)n_in; (void)out_size; (void)ws_size;
    const float* feats   = (const float*)d_in[0];   // mask_feats (2,8,128,192) f32
    const float* params  = (const float*)d_in[1];   // mask_head_params (128,169) f32
    const float* iloc    = (const float*)d_in[2];   // instance_locations (128,2) f32
    const int*   im_inds = (const int*)d_in[3];     // (128) i32
    const int*   fpn     = (const int*)d_in[4];     // (128) i32
    // d_in[5] = mask_feat_stride (scalar 8) -- geometry hardcoded to match

    float* logits = (float*)d_ws;                   // 128*24576 floats = 12.6 MB

    dim3 grid1(NINST, 12);
    dynmask_mlp_kernel<<<grid1, 256, 0, stream>>>(feats, params, iloc, im_inds, fpn, logits);

    const size_t total = (size_t)NINST * (2 * H) * (2 * W);
    const int blocks = (int)((total + 255) / 256);
    dynmask_upsample_kernel<<<blocks, 256, 0, stream>>>(logits, (float*)d_out);
}